// AerialPatchSampler_16020228014401
// MI455X (gfx1250) — compile-verified
//
#include <hip/hip_runtime.h>
#include <math.h>

#define HBp 64
#define WBp 64
#define HA 512
#define WA 512
#define CC 64
#define NP 128
#define HWSZ (HA*WA)
#define NPIX 4096

typedef __attribute__((ext_vector_type(2))) float v2f;
typedef __attribute__((ext_vector_type(8))) float v8f;

__global__ __launch_bounds__(256) void AerialPatchSampler_kernel(
    const float* __restrict__ aer,     // (2,64,512,512)
    const float* __restrict__ pose,    // (2,128,3)
    const float* __restrict__ oscale,  // (2,)
    float* __restrict__ out)           // (2,128,64,64,64)
{
  __shared__ float s_wx[NPIX];
  __shared__ float s_wy[NPIX];
  __shared__ int   s_pk[NPIX];

  const int bn   = blockIdx.x;          // b*NP + n
  const int cg   = blockIdx.y;          // channel group: 16 channels each
  const int b    = bn >> 7;
  const int t    = threadIdx.x;
  const int lane = t & 31;
  const int wave = t >> 5;

  const float u  = pose[bn*3 + 0];
  const float v  = pose[bn*3 + 1];
  const float th = pose[bn*3 + 2];
  const float os = oscale[b];
  const float cr = cosf(-th);
  const float sr = sinf(-th);

  // ---- Phase 1: grid transform via V_WMMA_F32_16X16X4_F32 ----
  // D(16x16) = A(16x4) x B(4x16):
  //   A row0 = [cr, -sr, u, 0], A row1 = [sr, cr, v, 0], rows 2..15 = 0
  //   B col n = [gu0(p), gv0(p), 1, 0]  (p = pixel for column n)
  // => D[0][n] = grid_u(p), D[1][n] = grid_v(p)   (exact, incl. translation)
  v2f amat; amat.x = 0.0f; amat.y = 0.0f;
  if      (lane == 0)  { amat.x = cr; amat.y = -sr; }  // M=0, K0/K1
  else if (lane == 1)  { amat.x = sr; amat.y =  cr; }  // M=1, K0/K1
  else if (lane == 16) { amat.x = u;  amat.y = 0.0f; } // M=0, K2/K3
  else if (lane == 17) { amat.x = v;  amat.y = 0.0f; } // M=1, K2/K3
  v8f cz = {};

  const int  n16 = lane & 15;
  const bool lo  = lane < 16;
  for (int it = 0; it < 32; ++it) {
    const int p = wave*512 + it*16 + n16;   // pixel handled by column n16
    const int i = p >> 6;                   // patch row
    const int j = p & 63;                   // patch col
    v2f bmat;
    bmat.x = lo ? (float)(63 - i) * os : 1.0f;   // gu0 | K=2 row of ones
    bmat.y = lo ? (float)(j - 32) * os : 0.0f;   // gv0 | K=3 row of zeros
    v8f d = __builtin_amdgcn_wmma_f32_16x16x4_f32(
        false, amat, false, bmat, (short)0, cz, false, false);
    if (lo) {
      const float gu = d[0];                // grid_u(p)
      const float gv = d[1];                // grid_v(p)
      float gx = (gu + 0.5f) * (2.0f / (float)WA) - 1.0f;
      float gy = (gv + 0.5f) * (2.0f / (float)HA) - 1.0f;
      const bool val = (fabsf(gx) < 1.0f) && (fabsf(gy) < 1.0f);
      gx = val ? gx : 2.0f;
      gy = val ? gy : 2.0f;
      const float x = (gx + 1.0f) * ((float)WA * 0.5f) - 0.5f;
      const float y = (gy + 1.0f) * ((float)HA * 0.5f) - 0.5f;
      const float x0f = floorf(x), y0f = floorf(y);
      const int x0 = (int)x0f, y0 = (int)y0f;
      s_wx[p] = x - x0f;
      s_wy[p] = y - y0f;
      s_pk[p] = (y0 << 16) | (x0 & 0xffff);   // both fit in int16
    }
  }
  __syncthreads();

  // ---- Phase 1.5: hoist per-pixel interp state into registers ----
  // Thread t owns pixels p = k*1024 + t*4 + q  (k=0..3, q=0..3) -> b128 stores.
  float w00[16], w10[16], w01[16], w11[16];
  int   o00[16], dx1[16], dyW[16];
#pragma unroll
  for (int kk = 0; kk < 16; ++kk) {
    const int k = kk >> 2, q = kk & 3;
    const int p = k*1024 + t*4 + q;
    const float wx = s_wx[p];
    const float wy = s_wy[p];
    const int   pk = s_pk[p];
    const int x0 = (int)(short)(pk & 0xffff);
    const int y0 = pk >> 16;
    const int x1 = x0 + 1, y1 = y0 + 1;
    const float mx0 = (x0 >= 0 && x0 < WA) ? 1.0f : 0.0f;
    const float mx1 = (x1 >= 0 && x1 < WA) ? 1.0f : 0.0f;
    const float my0 = (y0 >= 0 && y0 < HA) ? 1.0f : 0.0f;
    const float my1 = (y1 >= 0 && y1 < HA) ? 1.0f : 0.0f;
    const int xc0 = min(max(x0, 0), WA-1), xc1 = min(max(x1, 0), WA-1);
    const int yc0 = min(max(y0, 0), HA-1), yc1 = min(max(y1, 0), HA-1);
    const float owx = 1.0f - wx, owy = 1.0f - wy;
    w00[kk] = owx * owy * (mx0 * my0);
    w10[kk] = wx  * owy * (mx1 * my0);
    w01[kk] = owx * wy  * (mx0 * my1);
    w11[kk] = wx  * wy  * (mx1 * my1);
    o00[kk] = yc0 * WA + xc0;
    dx1[kk] = xc1 - xc0;           // 0 or 1
    dyW[kk] = (yc1 - yc0) * WA;    // 0 or 512
  }

  // ---- Phase 2: channel loop, pure gather + FMA, coalesced b128 stores ----
  const float* __restrict__ srcb = aer + (size_t)b * CC * HWSZ;
  float*       __restrict__ dstb = out + (size_t)bn * CC * NPIX;
  const int c0 = cg * 16;
  for (int c = c0; c < c0 + 16; ++c) {
    const float* __restrict__ src = srcb + (size_t)c * HWSZ;
    float*       __restrict__ dst = dstb + (size_t)c * NPIX;
    if (c + 1 < CC)  // pull next channel plane toward cache (global_prefetch_b8)
      __builtin_prefetch(srcb + (size_t)(c + 1) * HWSZ + o00[0], 0, 1);
#pragma unroll
    for (int k = 0; k < 4; ++k) {
      float rv[4];
#pragma unroll
      for (int q = 0; q < 4; ++q) {
        const int kk = k*4 + q;
        const int o  = o00[kk];
        const float v00 = src[o];
        const float v10 = src[o + dx1[kk]];
        const float v01 = src[o + dyW[kk]];
        const float v11 = src[o + dx1[kk] + dyW[kk]];
        rv[q] = v00*w00[kk] + v10*w10[kk] + v01*w01[kk] + v11*w11[kk];
      }
      float4 r = make_float4(rv[0], rv[1], rv[2], rv[3]);
      *(float4*)(dst + k*1024 + t*4) = r;
    }
  }
}

extern "C" void kernel_launch(void* const* d_in, const int* in_sizes, int n_in,
                              void* d_out, int out_size, void* d_ws, size_t ws_size,
                              hipStream_t stream) {
  const float* aer    = (const float*)d_in[0];  // 2*64*512*512
  const float* pose   = (const float*)d_in[1];  // 2*128*3
  const float* oscale = (const float*)d_in[2];  // 2
  float* out = (float*)d_out;                   // 2*128*64*64*64

  dim3 grid(2 * NP, 4);   // one block per (b,n) x 4 channel-groups
  dim3 block(256);        // 8 wave32 waves
  AerialPatchSampler_kernel<<<grid, block, 0, stream>>>(aer, pose, oscale, out);
}